// SEBlock_74612171866186
// MI455X (gfx1250) — compile-verified
//
#include <hip/hip_runtime.h>

typedef __attribute__((ext_vector_type(2))) float v2f;
typedef __attribute__((ext_vector_type(4))) float v4f;
typedef __attribute__((ext_vector_type(8))) float v8f;

#define BATCH 256
#define CH    512
#define SE    32
#define HW    1024   // 32*32

// ---------------------------------------------------------------------------
// Kernel 1: global average pool. One wave (32 lanes) per (b,c) plane of 1024
// floats; 8 waves per 256-thread block. float4 loads, wave32 xor-reduction.
// ---------------------------------------------------------------------------
__global__ __launch_bounds__(256) void se_pool(const float* __restrict__ x,
                                               float* __restrict__ pooled) {
  const int wave = threadIdx.x >> 5;
  const int lane = threadIdx.x & 31;
  const long plane = (long)blockIdx.x * 8 + wave;   // < 256*512
  const v4f* p = (const v4f*)(x + plane * HW);
  float s = 0.0f;
#pragma unroll
  for (int i = 0; i < 8; ++i) {
    v4f v = p[lane + 32 * i];
    s += v.x + v.y + v.z + v.w;
  }
#pragma unroll
  for (int off = 16; off > 0; off >>= 1)
    s += __shfl_xor(s, off, 32);
  if (lane == 0) pooled[plane] = s * (1.0f / (float)HW);
}

// ---------------------------------------------------------------------------
// Kernel 2: the SE MLP as two WMMA GEMMs (V_WMMA_F32_16X16X4_F32, fp32 exact).
// One wave per 16-row batch block; 16 blocks cover M=256.
//   GEMM1: pooled(16x512) x fc1_w^T(512x32) -> relu -> Y1(16x32)  (via LDS)
//   GEMM2: Y1(16x32) x fc2_w^T(32x1024) -> +b2 -> sigmoid/scale, bias planes
// ISA layouts (wave32):
//   A 16x4:  lane r(<16) holds A[r][k0],A[r][k0+1]; lanes 16-31 hold K+2,K+3
//   B 4x16:  VGPR0 = rows K (lanes 0-15) / K+2 (lanes 16-31), VGPR1 = K+1/K+3
//   C/D 16x16: VGPR v = row v (lanes 0-15) / row v+8 (lanes 16-31), col = lane%16
// ---------------------------------------------------------------------------
__global__ __launch_bounds__(32) void se_mlp(const float* __restrict__ pooled,
                                             const float* __restrict__ fc1_w,
                                             const float* __restrict__ fc1_b,
                                             const float* __restrict__ fc2_w,
                                             const float* __restrict__ fc2_b,
                                             float* __restrict__ scale,
                                             float* __restrict__ biasb) {
  __shared__ float y1[16 * SE];                 // 16 rows x 32 cols, 2 KB

  const int lane = threadIdx.x;                 // 0..31, full wave (EXEC all-1s)
  const int half = lane >> 4;                   // 0 or 1
  const int r    = lane & 15;
  const int m0   = blockIdx.x * 16;             // batch-row block

  // ---- GEMM1: K = 512 in steps of 4, two N-tiles (SE cols 0-15 and 16-31)
  const float* arow  = pooled + (size_t)(m0 + r) * CH + 2 * half;  // A row m0+r
  const float* b0row = fc1_w  + (size_t)(r)      * CH + 2 * half;  // B^T row n=r
  const float* b1row = fc1_w  + (size_t)(16 + r) * CH + 2 * half;  // B^T row n=16+r
  v8f acc0 = {};
  v8f acc1 = {};
  for (int k0 = 0; k0 < CH; k0 += 4) {
    v2f a  = *(const v2f*)(arow  + k0);
    v2f b0 = *(const v2f*)(b0row + k0);
    v2f b1 = *(const v2f*)(b1row + k0);
    acc0 = __builtin_amdgcn_wmma_f32_16x16x4_f32(false, a, false, b0,
                                                 (short)0, acc0, false, false);
    acc1 = __builtin_amdgcn_wmma_f32_16x16x4_f32(false, a, false, b1,
                                                 (short)0, acc1, false, false);
  }

  // bias + relu, stage Y1 to LDS (transposes D col-per-lane -> A row-per-lane)
  const float bb0 = fc1_b[r];
  const float bb1 = fc1_b[16 + r];
#pragma unroll
  for (int v = 0; v < 8; ++v) {
    y1[(v + 8 * half) * SE + r]      = fmaxf(acc0[v] + bb0, 0.0f);
    y1[(v + 8 * half) * SE + 16 + r] = fmaxf(acc1[v] + bb1, 0.0f);
  }
  __syncthreads();

  // ---- GEMM2: K = 32 in 8 steps, 64 N-tiles of 16 over the 1024 outputs
  const float* yrow = y1 + r * SE + 2 * half;   // A row r of Y1
  for (int nt = 0; nt < 64; ++nt) {
    const int n1 = nt * 16;
    const float* w2row = fc2_w + (size_t)(n1 + r) * SE + 2 * half; // B^T row n1+r
    v8f acc = {};
#pragma unroll
    for (int k0 = 0; k0 < SE; k0 += 4) {
      v2f a = *(const v2f*)(yrow  + k0);
      v2f b = *(const v2f*)(w2row + k0);
      acc = __builtin_amdgcn_wmma_f32_16x16x4_f32(false, a, false, b,
                                                  (short)0, acc, false, false);
    }
    const float bb = fc2_b[n1 + r];
    if (n1 < CH) {                                // scale half: sigmoid
#pragma unroll
      for (int v = 0; v < 8; ++v) {
        float t = acc[v] + bb;
        scale[(size_t)(m0 + v + 8 * half) * CH + n1 + r] =
            1.0f / (1.0f + __expf(-t));
      }
    } else {                                      // bias half: raw
#pragma unroll
      for (int v = 0; v < 8; ++v) {
        biasb[(size_t)(m0 + v + 8 * half) * CH + (n1 - CH) + r] = acc[v] + bb;
      }
    }
  }
}

// ---------------------------------------------------------------------------
// Kernel 3: out = x * scale[b,c] + bias[b,c]. One 256-thread block per plane,
// float4 FMA, nontemporal store (out is never re-read on device).
// ---------------------------------------------------------------------------
__global__ __launch_bounds__(256) void se_apply(const float* __restrict__ x,
                                                const float* __restrict__ scale,
                                                const float* __restrict__ biasb,
                                                float* __restrict__ out) {
  const long plane = blockIdx.x;                // < 256*512
  const float s = scale[plane];
  const float b = biasb[plane];
  const v4f* xp = (const v4f*)(x + plane * HW);
  v4f* op = (v4f*)(out + plane * HW);
  v4f v = xp[threadIdx.x];
  v4f r;
  r.x = fmaf(v.x, s, b);
  r.y = fmaf(v.y, s, b);
  r.z = fmaf(v.z, s, b);
  r.w = fmaf(v.w, s, b);
  __builtin_nontemporal_store(r, op + threadIdx.x);
}

// ---------------------------------------------------------------------------
extern "C" void kernel_launch(void* const* d_in, const int* in_sizes, int n_in,
                              void* d_out, int out_size, void* d_ws, size_t ws_size,
                              hipStream_t stream) {
  const float* x     = (const float*)d_in[0];   // (256, 512, 32, 32)
  const float* fc1_w = (const float*)d_in[1];   // (32, 512)
  const float* fc1_b = (const float*)d_in[2];   // (32,)
  const float* fc2_w = (const float*)d_in[3];   // (1024, 32)
  const float* fc2_b = (const float*)d_in[4];   // (1024,)
  float* out = (float*)d_out;

  float* pooled = (float*)d_ws;                 // 256*512 floats
  float* scale  = pooled + BATCH * CH;          // 256*512 floats
  float* biasb  = scale  + BATCH * CH;          // 256*512 floats  (1.5 MB total)

  const int planes = BATCH * CH;                // 131072

  se_pool<<<planes / 8, 256, 0, stream>>>(x, pooled);
  se_mlp<<<BATCH / 16, 32, 0, stream>>>(pooled, fc1_w, fc1_b, fc2_w, fc2_b,
                                        scale, biasb);
  se_apply<<<planes, 256, 0, stream>>>(x, scale, biasb, out);
}